// ALEError_23742579212666
// MI455X (gfx1250) — compile-verified
//
#include <hip/hip_runtime.h>

// loss = ALPHA * mean(sobel(target-pred)) + (1-ALPHA) * masked_mse
// mean(sobel(d)) collapses (linearity + telescoping with symmetric pad) to
//   128 * sum_rows(d[...,W-1] - d[...,0]) / N      (W = 1024, N = 32*512*1024)
// so everything is a single streaming pass over pred & target:
// 128 MB @ 23.3 TB/s  =>  ~5.5 us roofline floor.

#define ALE_ALPHA 0.3f
#define ALE_QPR   256   // float4 quads per row (W=1024)

typedef __attribute__((ext_vector_type(2))) float v2f;
typedef __attribute__((ext_vector_type(8))) float v8f;

// Full 32-lane wave sum via V_WMMA_F32_16X16X4_F32 with B = ones (B's VGPR
// layout is then irrelevant) and A in the documented 16x4 layout:
// lane L (<16):  A[L][0]=v, A[L][1]=0 ; lane L (>=16): A[L-16][2]=v, A[L-16][3]=0
// => D[m][n] = v_m + v_{m+16}. Summing D's 8 VGPRs leaves lanes 0-15 with
// sum(v0..7, v16..23) and lanes 16-31 with sum(v8..15, v24..31); one xor-16
// shuffle completes the reduction (result in every lane).
__device__ __forceinline__ float ale_wave_sum(float v) {
#if __has_builtin(__builtin_amdgcn_wmma_f32_16x16x4_f32)
    v2f a; a[0] = v;    a[1] = 0.0f;
    v2f b; b[0] = 1.0f; b[1] = 1.0f;
    v8f c = {};
    v8f d = __builtin_amdgcn_wmma_f32_16x16x4_f32(
        /*neg_a=*/false, a, /*neg_b=*/false, b,
        /*c_mod=*/(short)0, c, /*reuse_a=*/false, /*reuse_b=*/false);
    float t = ((d[0] + d[1]) + (d[2] + d[3])) + ((d[4] + d[5]) + (d[6] + d[7]));
    t += __shfl_xor(t, 16, 32);
    return t;
#else
    for (int m = 16; m > 0; m >>= 1) v += __shfl_xor(v, m, 32);
    return v;
#endif
}

__device__ __forceinline__ void ale_accum(float4 p, float4 t, int i,
                                          float& sumsq, float& cnt, float& edge) {
    float dx = t.x - p.x, dy = t.y - p.y, dz = t.z - p.z, dw = t.w - p.w;
    sumsq += (t.x > 0.0f) ? dx * dx : 0.0f;
    sumsq += (t.y > 0.0f) ? dy * dy : 0.0f;
    sumsq += (t.z > 0.0f) ? dz * dz : 0.0f;
    sumsq += (t.w > 0.0f) ? dw * dw : 0.0f;
    cnt   += (t.x > 0.0f) ? 1.0f : 0.0f;
    cnt   += (t.y > 0.0f) ? 1.0f : 0.0f;
    cnt   += (t.z > 0.0f) ? 1.0f : 0.0f;
    cnt   += (t.w > 0.0f) ? 1.0f : 0.0f;
    int col = i & (ALE_QPR - 1);                 // quad index within row
    edge += (col == 0)           ? -dx : 0.0f;   // w == 0
    edge += (col == ALE_QPR - 1) ?  dw : 0.0f;   // w == W-1
}

__global__ void ale_init(float* ws) {
    ws[0] = 0.0f;  // sum of valid diff^2
    ws[1] = 0.0f;  // valid count
    ws[2] = 0.0f;  // edge sum: sum_rows(d_last - d_first)
}

__global__ __launch_bounds__(256)
void ale_main(const float* __restrict__ pred, const float* __restrict__ tgt,
              float* __restrict__ ws, int nQuads) {
    const float4* __restrict__ p4 = (const float4*)pred;
    const float4* __restrict__ t4 = (const float4*)tgt;

    int tid    = blockIdx.x * blockDim.x + threadIdx.x;
    int stride = gridDim.x * blockDim.x;

    float sumsq = 0.0f, cnt = 0.0f, edge = 0.0f;

    int i = tid;
    // 4x-unrolled chunk: 8 independent global_load_b128 in flight per thread.
    for (; i + 3 * stride < nQuads; i += 4 * stride) {
        // Speculative prefetch of the next chunk (HBM -> GL2); out-of-range
        // addresses are silently dropped, so no bounds check / branch needed.
        __builtin_prefetch(&p4[i + 4 * stride], 0, 0);   // global_prefetch_b8
        __builtin_prefetch(&t4[i + 4 * stride], 0, 0);

        float4 pa = p4[i];
        float4 pb = p4[i + stride];
        float4 pc = p4[i + 2 * stride];
        float4 pd = p4[i + 3 * stride];
        float4 ta = t4[i];
        float4 tb = t4[i + stride];
        float4 tc = t4[i + 2 * stride];
        float4 td = t4[i + 3 * stride];

        ale_accum(pa, ta, i,              sumsq, cnt, edge);
        ale_accum(pb, tb, i + stride,     sumsq, cnt, edge);
        ale_accum(pc, tc, i + 2 * stride, sumsq, cnt, edge);
        ale_accum(pd, td, i + 3 * stride, sumsq, cnt, edge);
    }
    for (; i < nQuads; i += stride)       // generic tail
        ale_accum(p4[i], t4[i], i, sumsq, cnt, edge);

    // Wave reduction via WMMA (EXEC is all-ones here: no divergence).
    float s = ale_wave_sum(sumsq);
    float c = ale_wave_sum(cnt);
    float e = ale_wave_sum(edge);

    __shared__ float lds[8][3];                  // 256 threads = 8 waves
    int lane = threadIdx.x & 31;
    int wave = threadIdx.x >> 5;
    if (lane == 0) { lds[wave][0] = s; lds[wave][1] = c; lds[wave][2] = e; }
    __syncthreads();

    if (threadIdx.x == 0) {
        float S = 0.0f, C = 0.0f, E = 0.0f;
        #pragma unroll
        for (int w = 0; w < 8; ++w) { S += lds[w][0]; C += lds[w][1]; E += lds[w][2]; }
        atomicAdd(&ws[0], S);
        atomicAdd(&ws[1], C);
        atomicAdd(&ws[2], E);
    }
}

__global__ void ale_final(const float* __restrict__ ws, float* __restrict__ out,
                          float invN) {
    float sumsq = ws[0], cnt = ws[1], edge = ws[2];
    float mean_g = 128.0f * edge * invN;         // 4^3 smoothing * 2*(last-first)
    float mae    = sumsq / cnt;
    out[0] = ALE_ALPHA * mean_g + (1.0f - ALE_ALPHA) * mae;
}

extern "C" void kernel_launch(void* const* d_in, const int* in_sizes, int n_in,
                              void* d_out, int out_size, void* d_ws, size_t ws_size,
                              hipStream_t stream) {
    const float* pred = (const float*)d_in[0];
    const float* tgt  = (const float*)d_in[1];
    float*       out  = (float*)d_out;
    float*       ws   = (float*)d_ws;

    long long n = (long long)in_sizes[0];        // 32*1*512*1024 = 16777216
    int nQuads  = (int)(n / 4);

    ale_init<<<1, 1, 0, stream>>>(ws);
    // 2048 blocks: each thread runs the unrolled chunk twice, so the
    // next-chunk prefetch is genuinely useful on the first pass.
    ale_main<<<2048, 256, 0, stream>>>(pred, tgt, ws, nQuads);
    ale_final<<<1, 1, 0, stream>>>(ws, out, 1.0f / (float)n);
}